// ChamferLoss_28200755266074
// MI455X (gfx1250) — compile-verified
//
#include <hip/hip_runtime.h>

typedef float v2f __attribute__((ext_vector_type(2)));
typedef float v4f __attribute__((ext_vector_type(4)));
typedef float v8f __attribute__((ext_vector_type(8)));

#define WAVES_PER_BLOCK 8
#define BLOCK (WAVES_PER_BLOCK * 32)

// Prep: refAug[b*NR + j] = {-2*y0, -2*y1, -2*y2, yy}  (WMMA B-operand layout, 16B/point)
__global__ __launch_bounds__(256) void chamfer_prep_kernel(
    const float* __restrict__ pts, v4f* __restrict__ aug, int total)
{
    int i = blockIdx.x * 256 + threadIdx.x;
    if (i >= total) return;
    const float* p = pts + (size_t)i * 3;
    float y0 = p[0], y1 = p[1], y2 = p[2];
    v4f r;
    r.x = -2.0f * y0;
    r.y = -2.0f * y1;
    r.z = -2.0f * y2;
    r.w = y0 * y0 + y1 * y1 + y2 * y2;
    aug[i] = r;
}

// For each query point, min squared distance to all ref points.
// query: [B, NQ, 3] raw, refAug: [B, NR, 4] = {-2y, yy}, outMin: [B, NQ]
// Per wave: 16 query rows; per 16-col ref tile one v_wmma_f32_16x16x4_f32:
//   A[i][:] = {x0,x1,x2,1} (wave-constant)   B[:][j] = {-2y0,-2y1,-2y2,yy_j}   C[i][j] = xx_i
//   D = A*B + C = ||x_i - y_j||^2
__global__ __launch_bounds__(BLOCK) void chamfer_min_kernel(
    const float* __restrict__ query, const float* __restrict__ refAug,
    float* __restrict__ outMin, int NQ, int NR)
{
    const int lane      = threadIdx.x & 31;
    const int waveInBlk = threadIdx.x >> 5;
    const int wavesPerBatch = NQ >> 4;
    const int gw   = blockIdx.x * WAVES_PER_BLOCK + waveInBlk;
    const int b    = gw / wavesPerBatch;
    const int tile = gw % wavesPerBatch;
    const int rowbase = tile << 4;
    const int sub  = lane & 15;
    const int half = lane >> 4;

    // Own query point
    const float* qp = query + ((size_t)b * NQ + rowbase + sub) * 3;
    const float x0 = qp[0], x1 = qp[1], x2 = qp[2];
    const float xx = x0 * x0 + x1 * x1 + x2 * x2;

    // A 16x4: lanes 0-15 hold K=0,1; lanes 16-31 hold K=2,3
    v2f a;
    a.x = half ? x2   : x0;
    a.y = half ? 1.0f : x1;

    // C accumulator: VGPR v, half h -> row (v + 8h): broadcast xx from lane (v+8h) in sub-group
    v8f cxx;
    #pragma unroll
    for (int v = 0; v < 8; ++v)
        cxx[v] = __shfl(xx, v + 8 * half, 16);

    v8f rmin;
    #pragma unroll
    for (int v = 0; v < 8; ++v) rmin[v] = 3.0e38f;

    // B operand: lane reads 8 bytes of point (ct*16 + sub): half 0 -> {-2y0,-2y1}, half 1 -> {-2y2,yy}
    const float* rbase  = refAug + ((size_t)b * NR << 2) + (half << 1);
    const int    ntiles = NR >> 4;

    #pragma unroll 8
    for (int ct = 0; ct < ntiles; ++ct) {
        v2f bb = *(const v2f*)(rbase + (((ct << 4) + sub) << 2));

        v8f d = __builtin_amdgcn_wmma_f32_16x16x4_f32(
            false, a, false, bb, (short)0, cxx, false, false);

        #pragma unroll
        for (int v = 0; v < 8; ++v)
            rmin[v] = fminf(rmin[v], d[v]);
    }

    // Min-reduce across the 16 lanes of each half (columns), per accumulator VGPR (row)
    #pragma unroll
    for (int off = 8; off >= 1; off >>= 1) {
        #pragma unroll
        for (int v = 0; v < 8; ++v)
            rmin[v] = fminf(rmin[v], __shfl_xor(rmin[v], off, 16));
    }

    if (sub == 0) {
        float* o = outMin + (size_t)b * NQ + rowbase + half * 8;
        #pragma unroll
        for (int v = 0; v < 8; ++v) o[v] = rmin[v];
    }
}

// Sum both min arrays with their means' scaling -> scalar loss.
__global__ __launch_bounds__(1024) void chamfer_reduce_kernel(
    const float* __restrict__ minPred, const float* __restrict__ minGt,
    float* __restrict__ out, int nPred, int nGt)
{
    __shared__ float smem[32];
    const float invP = 1.0f / (float)nPred;
    const float invG = 1.0f / (float)nGt;

    float local = 0.0f;
    for (int i = threadIdx.x; i < nPred; i += 1024) local += minPred[i] * invP;
    for (int i = threadIdx.x; i < nGt;   i += 1024) local += minGt[i]  * invG;

    #pragma unroll
    for (int off = 16; off >= 1; off >>= 1) local += __shfl_xor(local, off, 32);

    const int lane = threadIdx.x & 31, wv = threadIdx.x >> 5;
    if (lane == 0) smem[wv] = local;
    __syncthreads();
    if (wv == 0) {
        float s = smem[lane];          // 32 waves exactly fill smem
        #pragma unroll
        for (int off = 16; off >= 1; off >>= 1) s += __shfl_xor(s, off, 32);
        if (lane == 0) out[0] = s;
    }
}

extern "C" void kernel_launch(void* const* d_in, const int* in_sizes, int n_in,
                              void* d_out, int out_size, void* d_ws, size_t ws_size,
                              hipStream_t stream)
{
    (void)in_sizes; (void)n_in; (void)out_size; (void)ws_size;
    const int B = 4, N = 8192, M = 8192;

    const float* preds = (const float*)d_in[0];   // [B, M, 3]
    const float* gts   = (const float*)d_in[1];   // [B, N, 3]
    float* out = (float*)d_out;

    // Workspace layout
    float* minPred  = (float*)d_ws;                        // [B*M]
    float* minGt    = minPred + (size_t)B * M;             // [B*N]
    v4f*   predsAug = (v4f*)(minGt + (size_t)B * N);       // [B*M] x 16B
    v4f*   gtsAug   = predsAug + (size_t)B * M;            // [B*N] x 16B

    // Prep augmented ref arrays
    chamfer_prep_kernel<<<(B * M + 255) / 256, 256, 0, stream>>>(preds, predsAug, B * M);
    chamfer_prep_kernel<<<(B * N + 255) / 256, 256, 0, stream>>>(gts, gtsAug, B * N);

    const int blocksP = (B * (M / 16)) / WAVES_PER_BLOCK;  // 256 blocks, 2048 waves
    const int blocksG = (B * (N / 16)) / WAVES_PER_BLOCK;

    // loss_1: for each pred, nearest gt  -> query=preds, ref=gts
    chamfer_min_kernel<<<blocksP, BLOCK, 0, stream>>>(preds, (const float*)gtsAug,   minPred, M, N);
    // loss_2: for each gt, nearest pred  -> query=gts, ref=preds
    chamfer_min_kernel<<<blocksG, BLOCK, 0, stream>>>(gts,   (const float*)predsAug, minGt,   N, M);

    chamfer_reduce_kernel<<<1, 1024, 0, stream>>>(minPred, minGt, out, B * M, B * N);
}